// VectorQuantizer_26551487824074
// MI455X (gfx1250) — compile-verified
//
#include <hip/hip_runtime.h>

typedef float vf2 __attribute__((ext_vector_type(2)));
typedef float vf8 __attribute__((ext_vector_type(8)));

#define DIMS   128
#define CHUNK  32          // codebook rows staged in LDS per buffer
#define LDS_STRIDE 132     // padded floats per row (bank-conflict-free)
#define LDS_ROW_BYTES (LDS_STRIDE * 4)   // 528, 16B aligned

// Issue async global->LDS copies for one codebook chunk (CHUNK rows x 128 f32).
// 256 threads, each issues 4 x B128 (16B) transfers. EXEC is all-ones here.
__device__ __forceinline__ void issue_chunk_async(const float* __restrict__ src,
                                                  float* __restrict__ ldsBase) {
  unsigned tid = threadIdx.x;
#pragma unroll
  for (int q = 0; q < (CHUNK * 32) / 256; ++q) {   // 32 segs/row * CHUNK rows
    unsigned item = tid + q * 256;
    unsigned row = item >> 5;        // 0..CHUNK-1
    unsigned seg = item & 31;        // 16B segment within 512B row
    unsigned long long g =
        (unsigned long long)((const char*)src + ((size_t)row << 9) + (seg << 4));
    unsigned l =
        (unsigned)(unsigned long long)((char*)ldsBase + row * LDS_ROW_BYTES + (seg << 4));
    asm volatile("global_load_async_to_lds_b128 %0, %1, off"
                 :: "v"(l), "v"(g) : "memory");
  }
}

__device__ __forceinline__ void wait_async_all() {
  asm volatile("s_wait_asynccnt 0x0" ::: "memory");
}

// ---------------------------------------------------------------------------
// Per-row squared norm + 1/max(norm, eps)  (one wave per 128-float row)
// ---------------------------------------------------------------------------
__global__ void rownorm_kernel(const float* __restrict__ X, int nrows,
                               float* __restrict__ sq, float* __restrict__ inv) {
  int row  = blockIdx.x * (blockDim.x >> 5) + (threadIdx.x >> 5);
  int lane = threadIdx.x & 31;
  if (row >= nrows) return;
  float4 v = ((const float4*)(X + (size_t)row * DIMS))[lane];
  float s = v.x * v.x + v.y * v.y + v.z * v.z + v.w * v.w;
  for (int off = 16; off; off >>= 1) s += __shfl_xor(s, off, 32);
  if (lane == 0) {
    sq[row]  = s;
    inv[row] = 1.0f / fmaxf(sqrtf(s), 1e-12f);
  }
}

// ---------------------------------------------------------------------------
// Init accumulators + histogram
// ---------------------------------------------------------------------------
__global__ void init_kernel(unsigned int* __restrict__ hist, int K,
                            double* __restrict__ accums,
                            unsigned int* __restrict__ cnt,
                            unsigned int* __restrict__ minbits) {
  int i = blockIdx.x * blockDim.x + threadIdx.x;
  if (i < K) hist[i] = 0u;
  if (i < 8) accums[i] = 0.0;
  if (i == 0) { *cnt = 0u; *minbits = 0x7F800000u; }  // +inf
}

// ---------------------------------------------------------------------------
// N x K distance argmin via fp32 WMMA.  8 waves/block, each wave = 16 tokens.
// Codebook double-buffered in LDS via async global->LDS loads.
// ---------------------------------------------------------------------------
__global__ __launch_bounds__(256) void argmin_kernel(
    const float* __restrict__ latent, const float* __restrict__ codebook,
    const float* __restrict__ xsq, const float* __restrict__ csq,
    int N, int K,
    int* __restrict__ bestIdx, float* __restrict__ bestDot) {
  __shared__ float ldsB[2][CHUNK * LDS_STRIDE];
  int lane = threadIdx.x & 31;
  int tokBase = (blockIdx.x * 8 + (threadIdx.x >> 5)) * 16;

  // A panel: 16 tokens x 128, WMMA 16x4 f32 layout, resident in registers.
  int t    = lane & 15;
  int dofs = (lane >> 4) * 2;
  vf2 a[32];
  {
    const float* rowp = latent + (size_t)(tokBase + t) * DIMS;
#pragma unroll
    for (int s = 0; s < 32; ++s) a[s] = *(const vf2*)(rowp + 4 * s + dofs);
  }

  float xs[8];
#pragma unroll
  for (int v = 0; v < 8; ++v) xs[v] = xsq[tokBase + v + ((lane >> 4) << 3)];

  float minv[8]; int mini[8]; float mind[8];
#pragma unroll
  for (int v = 0; v < 8; ++v) { minv[v] = 3.4e38f; mini[v] = 0; mind[v] = 0.f; }

  int nchunk = K / CHUNK;
  issue_chunk_async(codebook, ldsB[0]);
  for (int ch = 0; ch < nchunk; ++ch) {
    wait_async_all();        // our wave's loads for chunk ch are in LDS
    __syncthreads();         // everyone's loads done; prior reads retired
    if (ch + 1 < nchunk)
      issue_chunk_async(codebook + (size_t)(ch + 1) * CHUNK * DIMS,
                        ldsB[(ch + 1) & 1]);
    const float* buf = ldsB[ch & 1];

    for (int ct = 0; ct < CHUNK / 16; ++ct) {
      const float* brow = buf + (size_t)(ct * 16 + (lane & 15)) * LDS_STRIDE;
      vf8 c = {0.f, 0.f, 0.f, 0.f, 0.f, 0.f, 0.f, 0.f};
      // software-pipelined: preload 16 B slices, then chain 16 WMMAs, twice
#pragma unroll
      for (int h = 0; h < 2; ++h) {
        vf2 b[16];
#pragma unroll
        for (int s = 0; s < 16; ++s)
          b[s] = *(const vf2*)(brow + 4 * (h * 16 + s) + dofs);
#pragma unroll
        for (int s = 0; s < 16; ++s)
          c = __builtin_amdgcn_wmma_f32_16x16x4_f32(false, a[h * 16 + s], false,
                                                    b[s], (short)0, c, false, false);
      }
      int code = ch * CHUNK + ct * 16 + (lane & 15);
      float cs = csq[code];
#pragma unroll
      for (int v = 0; v < 8; ++v) {
        float dot = c[v];
        float d2  = xs[v] + cs - 2.0f * dot;
        if (d2 < minv[v]) { minv[v] = d2; mini[v] = code; mind[v] = dot; }
      }
    }
  }

  // reduce across the 16-lane half holding the same token's candidates
#pragma unroll
  for (int v = 0; v < 8; ++v) {
    float bv = minv[v]; int bi = mini[v]; float bd = mind[v];
    for (int off = 8; off; off >>= 1) {
      float ov = __shfl_xor(bv, off, 16);
      int   oi = __shfl_xor(bi, off, 16);
      float od = __shfl_xor(bd, off, 16);
      if (ov < bv || (ov == bv && oi < bi)) { bv = ov; bi = oi; bd = od; }
    }
    if ((lane & 15) == 0) {
      int tok = tokBase + v + ((lane >> 4) << 3);
      bestIdx[tok] = bi;
      bestDot[tok] = bd;
    }
  }
}

// ---------------------------------------------------------------------------
// Gather quantized rows, MSE, histogram, selected-cosine sum (1 wave / token)
// ---------------------------------------------------------------------------
__global__ void gather_kernel(const float* __restrict__ latent,
                              const float* __restrict__ codebook,
                              const int* __restrict__ bestIdx,
                              const float* __restrict__ bestDot,
                              const float* __restrict__ xinv,
                              const float* __restrict__ cinv, int N,
                              float* __restrict__ outQ,
                              float* __restrict__ outIdx,
                              unsigned int* __restrict__ hist,
                              double* __restrict__ mse_sum,
                              double* __restrict__ cos_sum) {
  int tok  = blockIdx.x * (blockDim.x >> 5) + (threadIdx.x >> 5);
  int lane = threadIdx.x & 31;
  if (tok >= N) return;
  int idx  = bestIdx[tok];
  float4 l = ((const float4*)(latent   + (size_t)tok * DIMS))[lane];
  float4 q = ((const float4*)(codebook + (size_t)idx * DIMS))[lane];
  ((float4*)(outQ + (size_t)tok * DIMS))[lane] = q;   // quantized_st == quantized (value)
  float dx = l.x - q.x, dy = l.y - q.y, dz = l.z - q.z, dw = l.w - q.w;
  float s = dx * dx + dy * dy + dz * dz + dw * dw;
  for (int off = 16; off; off >>= 1) s += __shfl_xor(s, off, 32);
  if (lane == 0) {
    outIdx[tok] = (float)idx;
    atomicAdd(&hist[idx], 1u);
    atomicAdd(mse_sum, (double)s);
    atomicAdd(cos_sum, (double)(bestDot[tok] * xinv[tok] * cinv[idx]));
  }
}

// ---------------------------------------------------------------------------
// K x K codebook diversity: one fp32 WMMA GEMM, penalties reduced on the fly
// ---------------------------------------------------------------------------
__global__ __launch_bounds__(256) void diversity_kernel(
    const float* __restrict__ codebook, const float* __restrict__ csq,
    const float* __restrict__ cinv, int K,
    double* __restrict__ sim_sum, double* __restrict__ relu_sum,
    double* __restrict__ d_sum, unsigned int* __restrict__ cnt_sum,
    unsigned int* __restrict__ minbits) {
  __shared__ float ldsB[2][CHUNK * LDS_STRIDE];
  int lane = threadIdx.x & 31;
  int iBase = (blockIdx.x * 8 + (threadIdx.x >> 5)) * 16;

  int t    = lane & 15;
  int dofs = (lane >> 4) * 2;
  vf2 a[32];
  {
    const float* rowp = codebook + (size_t)(iBase + t) * DIMS;
#pragma unroll
    for (int s = 0; s < 32; ++s) a[s] = *(const vf2*)(rowp + 4 * s + dofs);
  }
  float ci_sq[8], ci_inv[8];
#pragma unroll
  for (int v = 0; v < 8; ++v) {
    int i = iBase + v + ((lane >> 4) << 3);
    ci_sq[v] = csq[i]; ci_inv[v] = cinv[i];
  }

  double l_sim = 0.0, l_relu = 0.0, l_d = 0.0;
  unsigned int l_cnt = 0;
  float l_min = __int_as_float(0x7F800000);

  int nchunk = K / CHUNK;
  issue_chunk_async(codebook, ldsB[0]);
  for (int ch = 0; ch < nchunk; ++ch) {
    wait_async_all();
    __syncthreads();
    if (ch + 1 < nchunk)
      issue_chunk_async(codebook + (size_t)(ch + 1) * CHUNK * DIMS,
                        ldsB[(ch + 1) & 1]);
    const float* buf = ldsB[ch & 1];

    for (int ct = 0; ct < CHUNK / 16; ++ct) {
      const float* brow = buf + (size_t)(ct * 16 + (lane & 15)) * LDS_STRIDE;
      vf8 c = {0.f, 0.f, 0.f, 0.f, 0.f, 0.f, 0.f, 0.f};
#pragma unroll
      for (int h = 0; h < 2; ++h) {
        vf2 b[16];
#pragma unroll
        for (int s = 0; s < 16; ++s)
          b[s] = *(const vf2*)(brow + 4 * (h * 16 + s) + dofs);
#pragma unroll
        for (int s = 0; s < 16; ++s)
          c = __builtin_amdgcn_wmma_f32_16x16x4_f32(false, a[h * 16 + s], false,
                                                    b[s], (short)0, c, false, false);
      }
      int j = ch * CHUNK + ct * 16 + (lane & 15);
      float cj_sq = csq[j], cj_inv = cinv[j];
#pragma unroll
      for (int v = 0; v < 8; ++v) {
        int i = iBase + v + ((lane >> 4) << 3);
        float dot = c[v];
        float sim, d;
        if (i == j) { sim = 0.f; d = 0.f; }
        else {
          sim = dot * ci_inv[v] * cj_inv;
          float pd2 = fmaxf(ci_sq[v] + cj_sq - 2.0f * dot, 0.f);
          d = (pd2 > 0.f) ? sqrtf(pd2) : 0.f;
        }
        float cl = fminf(fmaxf(sim, -10.f), 10.f);
        l_sim  += (double)__expf(cl);
        l_relu += (double)fmaxf(2.0f - d, 0.f);
        if (d > 0.f) { l_d += (double)d; l_cnt++; l_min = fminf(l_min, d); }
      }
    }
  }

  for (int off = 16; off; off >>= 1) {
    l_sim  += __shfl_xor(l_sim, off, 32);
    l_relu += __shfl_xor(l_relu, off, 32);
    l_d    += __shfl_xor(l_d, off, 32);
    l_cnt  += __shfl_xor(l_cnt, off, 32);
    l_min   = fminf(l_min, __shfl_xor(l_min, off, 32));
  }
  if (lane == 0) {
    atomicAdd(sim_sum, l_sim);
    atomicAdd(relu_sum, l_relu);
    atomicAdd(d_sum, l_d);
    atomicAdd(cnt_sum, l_cnt);
    atomicMin(minbits, __float_as_uint(l_min));
  }
}

// ---------------------------------------------------------------------------
// Final scalars: losses, perplexity, averages
// ---------------------------------------------------------------------------
__global__ void finalize_kernel(const unsigned int* __restrict__ hist,
                                int N, int K,
                                const double* __restrict__ accums,
                                const unsigned int* __restrict__ cnt,
                                const unsigned int* __restrict__ minbits,
                                float* __restrict__ outS) {
  __shared__ double sh[256];
  double h = 0.0;
  for (int k = threadIdx.x; k < K; k += 256) {
    double p = (double)hist[k] / (double)N;
    h += p * log(p + 1e-10);
  }
  sh[threadIdx.x] = h;
  __syncthreads();
  for (int s = 128; s; s >>= 1) {
    if (threadIdx.x < s) sh[threadIdx.x] += sh[threadIdx.x + s];
    __syncthreads();
  }
  if (threadIdx.x == 0) {
    double mse     = accums[0] / ((double)N * (double)DIMS);
    double kk      = (double)K * (double)K;
    double simPen  = accums[2] / kk;
    double distPen = accums[3] / kk;
    double c       = *cnt ? (double)(*cnt) : 1.0;
    outS[0] = (float)(0.25 * mse);                        // commitment_loss
    outS[1] = (float)(mse + simPen + 0.2 * distPen);      // codebook + diversity
    outS[2] = (float)exp(-sh[0]);                         // perplexity
    outS[3] = (float)(accums[1] / (double)N);             // sel_cos
    outS[4] = (float)(accums[4] / c);                     // avg_euclidean
    outS[5] = __uint_as_float(*minbits);                  // min_euclidean
  }
}

// ---------------------------------------------------------------------------
extern "C" void kernel_launch(void* const* d_in, const int* in_sizes, int n_in,
                              void* d_out, int out_size, void* d_ws, size_t ws_size,
                              hipStream_t stream) {
  const float* latent   = (const float*)d_in[0];
  const float* codebook = (const float*)d_in[1];
  int N = in_sizes[0] / DIMS;   // 16384
  int K = in_sizes[1] / DIMS;   // 8192

  char* ws = (char*)d_ws;
  double*       accums  = (double*)ws;                    // [0]mse [1]cos [2]sim [3]relu [4]dsum
  unsigned int* cnt     = (unsigned int*)(ws + 64);
  unsigned int* minbits = cnt + 1;
  float* xsq  = (float*)(ws + 80);
  float* xinv = xsq + N;
  float* csq  = xinv + N;
  float* cinv = csq + K;
  int*   bestIdx = (int*)(cinv + K);
  float* bestDot = (float*)(bestIdx + N);
  unsigned int* hist = (unsigned int*)(bestDot + N);

  float* outQ    = (float*)d_out;
  float* outIdx  = outQ + (size_t)N * DIMS;
  float* outScal = outIdx + N;

  init_kernel<<<(K + 255) / 256, 256, 0, stream>>>(hist, K, accums, cnt, minbits);
  rownorm_kernel<<<N / 8, 256, 0, stream>>>(latent, N, xsq, xinv);
  rownorm_kernel<<<K / 8, 256, 0, stream>>>(codebook, K, csq, cinv);
  argmin_kernel<<<N / 128, 256, 0, stream>>>(latent, codebook, xsq, csq, N, K,
                                             bestIdx, bestDot);
  gather_kernel<<<N / 8, 256, 0, stream>>>(latent, codebook, bestIdx, bestDot,
                                           xinv, cinv, N, outQ, outIdx, hist,
                                           &accums[0], &accums[1]);
  diversity_kernel<<<K / 128, 256, 0, stream>>>(codebook, csq, cinv, K,
                                                &accums[2], &accums[3], &accums[4],
                                                cnt, minbits);
  finalize_kernel<<<1, 256, 0, stream>>>(hist, N, K, accums, cnt, minbits, outScal);
}